// NodeLinear_16088947491453
// MI455X (gfx1250) — compile-verified
//
#include <hip/hip_runtime.h>

#define NN 100000
#define NE 1600000
#define NG 50
#define FD 128

typedef __attribute__((ext_vector_type(2))) float v2f;
typedef __attribute__((ext_vector_type(8))) float v8f;

// ---------------- kernel 1: zero accumulators in workspace ----------------
__global__ void zero_f4_kernel(float4* __restrict__ p, long n4) {
    long i = (long)blockIdx.x * blockDim.x + threadIdx.x;
    long stride = (long)gridDim.x * blockDim.x;
    float4 z = make_float4(0.f, 0.f, 0.f, 0.f);
    for (; i < n4; i += stride) p[i] = z;
}

// ---------------- kernel 2: edge scatter (segment sums) ----------------
// one thread per (edge, feature); fp32 atomics land in L2 (accumulators fit)
__global__ __launch_bounds__(256) void edge_scatter_kernel(
    const float* __restrict__ ef, const int* __restrict__ recv,
    const int* __restrict__ send, float* __restrict__ agg_in,
    float* __restrict__ agg_out) {
    long tid = (long)blockIdx.x * 256 + threadIdx.x;
    if (tid >= (long)NE * FD) return;
    int e = (int)(tid >> 7);
    int f = (int)(tid & (FD - 1));
    float v = ef[tid];
    int r = recv[e];
    int s = send[e];
    unsafeAtomicAdd(&agg_in[(long)r * FD + f], v);
    unsafeAtomicAdd(&agg_out[(long)s * FD + f], v);
}

// ---------------- kernel 3: glob_bias[g][o] = G[g] . Wg[o] + bias[o] ----------------
__global__ void glob_proj_kernel(const float* __restrict__ G,
                                 const float* __restrict__ Wg,
                                 const float* __restrict__ bias,
                                 float* __restrict__ gb) {
    int g = blockIdx.x;
    int o = threadIdx.x;
    const float* grow = G + (long)g * FD;
    const float* wrow = Wg + (long)o * FD;
    float acc = 0.f;
#pragma unroll 8
    for (int k = 0; k < FD; ++k) acc += grow[k] * wrow[k];
    gb[(long)g * FD + o] = acc + bias[o];
}

// ---------------- kernel 4: fused triple GEMM + gather epilogue (WMMA f32) ----------------
// out[16x128 tile] = X@Wn^T + Ain@Wi^T + Aout@Wo^T + gb[graph_idx] ; exact fp32 via
// V_WMMA_F32_16X16X4_F32. One wave per 16-row stripe, 8 col-tiles of 16.
__global__ __launch_bounds__(256) void node_wmma_kernel(
    const float* __restrict__ X,   const float* __restrict__ Wn,
    const float* __restrict__ Ain, const float* __restrict__ Wi,
    const float* __restrict__ Aout,const float* __restrict__ Wo,
    const float* __restrict__ gb,  const int* __restrict__ gidx,
    float* __restrict__ out) {
    const int wave = threadIdx.x >> 5;
    const int lane = threadIdx.x & 31;
    const int tile = blockIdx.x * 8 + wave;
    if (tile >= NN / 16) return;          // wave-uniform: EXEC stays all-ones
    const int r0 = tile * 16;
    const int hi = lane >> 4;             // 0: lanes 0-15, 1: lanes 16-31
    const int lo = lane & 15;

    v8f acc[8];
#pragma unroll
    for (int t = 0; t < 8; ++t) acc[t] = (v8f)0.f;

    // A-matrix per-lane base: row = r0+lo, k offset = 2*hi  (16x4 f32 layout)
    const long aoff = (long)(r0 + lo) * FD + 2 * hi;
    const float* arN = X + aoff;
    const float* arI = Ain + aoff;
    const float* arO = Aout + aoff;

    for (int kk = 0; kk < FD; kk += 4) {
        v2f an = *(const v2f*)(arN + kk);
        v2f ai = *(const v2f*)(arI + kk);
        v2f ao = *(const v2f*)(arO + kk);
#pragma unroll
        for (int t = 0; t < 8; ++t) {
            // B = W^T tile: B[k][n] = W[n][k]; per-lane float2 at W + col*FD + kk + 2*hi
            const long bo = (long)(t * 16 + lo) * FD + 2 * hi + kk;
            v2f bn = *(const v2f*)(Wn + bo);
            v2f bi = *(const v2f*)(Wi + bo);
            v2f bw = *(const v2f*)(Wo + bo);
            acc[t] = __builtin_amdgcn_wmma_f32_16x16x4_f32(
                false, an, false, bn, (short)0, acc[t], false, false);
            acc[t] = __builtin_amdgcn_wmma_f32_16x16x4_f32(
                false, ai, false, bi, (short)0, acc[t], false, false);
            acc[t] = __builtin_amdgcn_wmma_f32_16x16x4_f32(
                false, ao, false, bw, (short)0, acc[t], false, false);
        }
    }

    // Epilogue: D VGPR r holds rows r0+r (lanes 0-15) / r0+r+8 (lanes 16-31)
#pragma unroll
    for (int r = 0; r < 8; ++r) {
        const int row = r0 + r + 8 * hi;
        const int g = gidx[row];
        const float* gbrow = gb + (long)g * FD;
        float* orow = out + (long)row * FD;
#pragma unroll
        for (int t = 0; t < 8; ++t) {
            const int col = t * 16 + lo;
            orow[col] = acc[t][r] + gbrow[col];
        }
    }
}

extern "C" void kernel_launch(void* const* d_in, const int* in_sizes, int n_in,
                              void* d_out, int out_size, void* d_ws, size_t ws_size,
                              hipStream_t stream) {
    const float* node_features   = (const float*)d_in[0];
    const float* edge_features   = (const float*)d_in[1];
    const float* global_features = (const float*)d_in[2];
    const float* W_node          = (const float*)d_in[3];
    const float* W_incoming      = (const float*)d_in[4];
    const float* W_outgoing      = (const float*)d_in[5];
    const float* W_global        = (const float*)d_in[6];
    const float* bias            = (const float*)d_in[7];
    const int*   receivers       = (const int*)d_in[8];
    const int*   senders         = (const int*)d_in[9];
    const int*   graph_idx       = (const int*)d_in[10];
    float* out = (float*)d_out;

    // workspace layout: agg_in | agg_out | glob_bias
    float* agg_in  = (float*)d_ws;
    float* agg_out = agg_in + (long)NN * FD;
    float* gb      = agg_out + (long)NN * FD;

    // 1) zero both accumulators (deterministic per call)
    const long n4 = (2L * NN * FD) / 4;
    zero_f4_kernel<<<4096, 256, 0, stream>>>((float4*)d_ws, n4);

    // 2) scatter edges into agg_in/agg_out
    const long ethreads = (long)NE * FD;
    const int eblocks = (int)((ethreads + 255) / 256);   // 800,000
    edge_scatter_kernel<<<eblocks, 256, 0, stream>>>(edge_features, receivers,
                                                     senders, agg_in, agg_out);

    // 3) glob_bias = global_features @ W_global^T + bias
    glob_proj_kernel<<<NG, FD, 0, stream>>>(global_features, W_global, bias, gb);

    // 4) fused WMMA GEMMs + gather epilogue
    const int ntiles = NN / 16;                          // 6250
    const int nblocks = (ntiles + 7) / 8;                // 782
    node_wmma_kernel<<<nblocks, 256, 0, stream>>>(node_features, W_node,
                                                  agg_in, W_incoming,
                                                  agg_out, W_outgoing,
                                                  gb, graph_idx, out);
}